// StandardROIHeads_85658827751896
// MI455X (gfx1250) — compile-verified
//
#include <hip/hip_runtime.h>
#include <hip/hip_bf16.h>
#include <stdint.h>

#define BIMG   32
#define MGT    256
#define NPROP  4000
#define KCOL   (NPROP + MGT)   // 4256
#define QCOL   (KCOL / 4)      // 1064 quad-columns
#define NCLS   80
#define BATCH  512
#define FG_CAP 128             // 512 * 0.25

typedef uint32_t u32x4 __attribute__((ext_vector_type(4)));
typedef uint32_t u32x8 __attribute__((ext_vector_type(8)));

static __device__ __forceinline__ unsigned int lds_addr_of(const void* p) {
  // generic pointer to a __shared__ object: low 32 bits are the LDS offset
  return (unsigned int)(unsigned long long)(uintptr_t)p;
}

static __device__ __forceinline__ void async_b128_to_lds(unsigned int lds_off,
                                                         const void* gaddr) {
  unsigned long long g = (unsigned long long)(uintptr_t)gaddr;
  asm volatile("global_load_async_to_lds_b128 %0, %1, off"
               :: "v"(lds_off), "v"(g) : "memory");
}

static __device__ __forceinline__ void wait_asynccnt0() {
  asm volatile("s_wait_asynccnt 0" ::: "memory");
}

// TDM: DMA `nelems` contiguous f32 from global to LDS (1-row tile).
// Issued once per executing wave; caller must ensure only one wave runs it,
// then s_wait_tensorcnt + barrier before consumers read LDS.
static __device__ __forceinline__ void tdm_load_f32_row(unsigned int lds_off,
                                                        const void* gaddr,
                                                        unsigned int nelems) {
  const unsigned long long ga = (unsigned long long)(uintptr_t)gaddr;
  u32x4 g0;
  g0[0] = 1u;                                   // count=1, user descriptor
  g0[1] = lds_off;                              // lds_addr [63:32]
  g0[2] = (uint32_t)ga;                         // global_addr[31:0]
  g0[3] = (uint32_t)(ga >> 32) | (2u << 30);    // global_addr[56:32] | type=2
  u32x8 g1;
  g1[0] = 2u << 16;                             // data_size=2 (4 bytes)
  g1[1] = (nelems & 0xFFFFu) << 16;             // tensor_dim0[15:0]
  g1[2] = (nelems >> 16) | (1u << 16);          // tensor_dim0[31:16] | tensor_dim1=1
  g1[3] = (nelems & 0xFFFFu) << 16;             // tile_dim0
  g1[4] = 1u;                                   // tile_dim1=1, tile_dim2=0
  g1[5] = nelems;                               // tensor_dim0_stride[31:0]
  g1[6] = 0u;                                   // stride hi | dim1_stride lo
  g1[7] = 0u;
  asm volatile("tensor_load_to_lds %0, %1" :: "s"(g0), "s"(g1) : "memory");
  __builtin_amdgcn_s_wait_tensorcnt(0);
}

// -------------------------------------------------------------------------
// Kernel 1: IoU matrix + per-column argmax/match + class assignment.
// grid = (ceil(1064/256)=5, B), block = 256 (8 wave32); 4 columns/thread,
// b128 stores. gt boxes staged to LDS via the Tensor Data Mover.
// -------------------------------------------------------------------------
__global__ __launch_bounds__(256) void iou_match_kernel(
    const float* __restrict__ gt_boxes,     // [B, 256, 4]
    const float* __restrict__ prop_boxes,   // [B, 4000, 4]
    const int*   __restrict__ gt_classes,   // [B, 256]
    float* __restrict__ out_iou,            // [B, 256, 4256]
    float* __restrict__ out_midx,           // [B, 4256]
    float* __restrict__ out_mlab,           // [B, 4256]
    unsigned char* __restrict__ ws_cls)     // [B, 4256]
{
  __shared__ __attribute__((aligned(16))) float4 sGT[MGT];   // 4 KB
  __shared__ float sArea[MGT];

  const int b = blockIdx.y;
  const int t = threadIdx.x;

  // Wave 0 issues a single TDM DMA of this image's gt boxes (1024 f32).
  if (t < 32)
    tdm_load_f32_row(lds_addr_of(&sGT[0]),
                     gt_boxes + (size_t)b * MGT * 4, MGT * 4);
  __syncthreads();

  // Precompute gt areas (one per thread) in LDS.
  {
    const float4 g = sGT[t];
    sArea[t] = (g.z - g.x) * (g.w - g.y);
  }
  __syncthreads();

  const int  q      = blockIdx.x * 256 + t;  // quad-column index
  const bool active = (q < QCOL);
  const int  j0     = q * 4;                 // first of 4 owned columns

  float4 P[4];
  float  pA[4];
#pragma unroll
  for (int c = 0; c < 4; ++c) P[c] = make_float4(0.f, 0.f, 0.f, 0.f);
  if (active) {
    if (j0 < NPROP) {          // quads never straddle (4000 % 4 == 0)
      const float4* pb = (const float4*)(prop_boxes + ((size_t)b * NPROP + j0) * 4);
#pragma unroll
      for (int c = 0; c < 4; ++c) P[c] = pb[c];
    } else {
#pragma unroll
      for (int c = 0; c < 4; ++c) P[c] = sGT[j0 - NPROP + c];
    }
  }
#pragma unroll
  for (int c = 0; c < 4; ++c) pA[c] = (P[c].z - P[c].x) * (P[c].w - P[c].y);

  float best[4]  = {-1.f, -1.f, -1.f, -1.f};
  int   bestm[4] = {0, 0, 0, 0};
  float* iou_row = out_iou + (size_t)b * MGT * KCOL + (size_t)j0;

#pragma unroll 2
  for (int m = 0; m < MGT; ++m) {
    const float4 g  = sGT[m];       // LDS broadcast (all lanes same addr)
    const float  ga = sArea[m];
    float4 r;
    float* rp = &r.x;
#pragma unroll
    for (int c = 0; c < 4; ++c) {
      const float lx = fmaxf(g.x, P[c].x), ly = fmaxf(g.y, P[c].y);
      const float rx = fminf(g.z, P[c].z), ry = fminf(g.w, P[c].w);
      const float w  = fmaxf(rx - lx, 0.0f), h = fmaxf(ry - ly, 0.0f);
      const float inter = w * h;
      const float uni   = ga + pA[c] - inter;
      // rcp (TRANS, co-executes) instead of IEEE div: keeps VALU/byte below
      // the HBM roofline (writes are the binding constraint at 23.3 TB/s).
      const float iou = (inter > 0.0f) ? inter * __builtin_amdgcn_rcpf(uni) : 0.0f;
      rp[c] = iou;
      if (iou > best[c]) { best[c] = iou; bestm[c] = m; }  // first-max argmax
    }
    if (active) *(float4*)(iou_row + (size_t)m * KCOL) = r;  // coalesced b128
  }

  if (active) {
    float4 midx, mlab;
    float* mi = &midx.x; float* ml = &mlab.x;
    unsigned char cls[4];
#pragma unroll
    for (int c = 0; c < 4; ++c) {
      const int lab = (best[c] >= 0.5f) ? 1 : 0;
      cls[c] = (unsigned char)(lab ? gt_classes[b * MGT + bestm[c]] : NCLS);
      mi[c] = (float)bestm[c];
      ml[c] = (float)lab;
    }
    *(float4*)(out_midx + (size_t)b * KCOL + j0) = midx;
    *(float4*)(out_mlab + (size_t)b * KCOL + j0) = mlab;
    *(uchar4*)(ws_cls + (size_t)b * KCOL + j0) =
        make_uchar4(cls[0], cls[1], cls[2], cls[3]);
  }
}

// -------------------------------------------------------------------------
// Kernel 2: rank-based fg/bg subsample + stable top-512 gather.
// grid = B, block = 1024 (32 wave32). ~26 KB LDS. Priorities staged via
// per-lane async global->LDS loads (ASYNCcnt).
// -------------------------------------------------------------------------
__global__ __launch_bounds__(1024) void subsample_kernel(
    const float* __restrict__ pri,          // [B, 4256]
    const unsigned char* __restrict__ ws_cls,
    float* __restrict__ out_idx,            // [B, 512]
    float* __restrict__ out_lab)            // [B, 512]
{
  __shared__ __attribute__((aligned(16))) float sPri[KCOL];
  __shared__ unsigned char sCls[KCOL];
  __shared__ unsigned char sSel[KCOL];
  __shared__ int sFg;

  const int b = blockIdx.x;
  const int t = threadIdx.x;

  // Async-stage priorities (17 KB) into LDS: 1064 b128 transfers.
  for (int u = t; u < KCOL / 4; u += 1024)
    async_b128_to_lds(lds_addr_of(&sPri[u * 4]),
                      pri + (size_t)b * KCOL + (size_t)u * 4);

  if (t == 0) sFg = 0;

  // Stage classes + local fg count (overlaps with async loads in flight).
  int localFg = 0;
#pragma unroll
  for (int k = 0; k < 5; ++k) {
    const int i = t + k * 1024;
    if (i < KCOL) {
      const unsigned char c = ws_cls[(size_t)b * KCOL + i];
      sCls[i] = c;
      localFg += (c < NCLS) ? 1 : 0;
    }
  }
  wait_asynccnt0();
  __syncthreads();
  if (localFg) atomicAdd(&sFg, localFg);
  __syncthreads();

  const int cntFg   = sFg;
  const int num_fg  = min(cntFg, FG_CAP);
  const int num_bg  = min(KCOL - cntFg, BATCH - num_fg);
  const int num_sel = num_fg + num_bg;     // <= 512 always

  // Each thread owns up to 5 elements.
  int ik[5]; bool vk[5]; bool fgk[5]; float pk[5]; int cnt[5];
#pragma unroll
  for (int k = 0; k < 5; ++k) {
    ik[k]  = t + k * 1024;
    vk[k]  = ik[k] < KCOL;
    pk[k]  = vk[k] ? sPri[ik[k]] : 0.0f;
    fgk[k] = vk[k] ? (sCls[ik[k]] < NCLS) : false;
    cnt[k] = 0;
  }

  // Stable descending rank within own group (all lanes read the same LDS
  // word per iteration -> broadcast, conflict-free).
  for (int jj = 0; jj < KCOL; ++jj) {
    const float pj  = sPri[jj];
    const bool  fgj = sCls[jj] < NCLS;
#pragma unroll
    for (int k = 0; k < 5; ++k) {
      const bool before = (pj > pk[k]) || (pj == pk[k] && jj < ik[k]);
      if (fgj == fgk[k] && before) cnt[k]++;
    }
  }
#pragma unroll
  for (int k = 0; k < 5; ++k)
    if (vk[k]) sSel[ik[k]] = (cnt[k] < (fgk[k] ? num_fg : num_bg)) ? 1 : 0;
  __syncthreads();

  // Stable top-512 placement: selected ordered by priority desc (ties by
  // index asc); padding = unselected in ascending index order (lax.top_k
  // stability on equal -inf keys).
  int rSel[5], rPad[5];
#pragma unroll
  for (int k = 0; k < 5; ++k) { rSel[k] = 0; rPad[k] = 0; }
  for (int jj = 0; jj < KCOL; ++jj) {
    const bool  sj = sSel[jj] != 0;
    const float pj = sPri[jj];
#pragma unroll
    for (int k = 0; k < 5; ++k) {
      const bool before = (pj > pk[k]) || (pj == pk[k] && jj < ik[k]);
      if (sj && before)       rSel[k]++;
      if (!sj && jj < ik[k])  rPad[k]++;
    }
  }
#pragma unroll
  for (int k = 0; k < 5; ++k) {
    if (!vk[k]) continue;
    if (sSel[ik[k]]) {
      const int r = rSel[k];                       // < num_sel <= 512
      out_idx[(size_t)b * BATCH + r] = (float)ik[k];
      out_lab[(size_t)b * BATCH + r] = (float)sCls[ik[k]];
    } else {
      const int pos = num_sel + rPad[k];
      if (pos < BATCH) {
        out_idx[(size_t)b * BATCH + pos] = (float)ik[k];
        out_lab[(size_t)b * BATCH + pos] = -1.0f;
      }
    }
  }
}

// -------------------------------------------------------------------------
extern "C" void kernel_launch(void* const* d_in, const int* in_sizes, int n_in,
                              void* d_out, int out_size, void* d_ws, size_t ws_size,
                              hipStream_t stream) {
  const float* gt_boxes   = (const float*)d_in[0];
  const float* prop_boxes = (const float*)d_in[1];
  const int*   gt_classes = (const int*)d_in[2];
  const float* pri        = (const float*)d_in[3];

  float* out = (float*)d_out;
  float* out_iou  = out;                                         // 34,865,152
  float* out_midx = out_iou  + (size_t)BIMG * MGT * KCOL;        //    136,192
  float* out_mlab = out_midx + (size_t)BIMG * KCOL;              //    136,192
  float* out_idx  = out_mlab + (size_t)BIMG * KCOL;              //     16,384
  float* out_lab  = out_idx  + (size_t)BIMG * BATCH;             //     16,384

  unsigned char* ws_cls = (unsigned char*)d_ws;                  // B*K bytes

  dim3 grid1((QCOL + 255) / 256, BIMG);
  iou_match_kernel<<<grid1, 256, 0, stream>>>(
      gt_boxes, prop_boxes, gt_classes, out_iou, out_midx, out_mlab, ws_cls);

  subsample_kernel<<<BIMG, 1024, 0, stream>>>(pri, ws_cls, out_idx, out_lab);
}